// NeuronGCN_2740189135610
// MI455X (gfx1250) — compile-verified
//
#include <hip/hip_runtime.h>
#include <hip/hip_bf16.h>

#define N_NODES 100000
#define N_EDGES 800000
#define C_IN    128
#define HID     64
#define HEADS   2
#define B_GR    64
#define OUT_DIM 10
#define EPSV    1e-5f
#define NSLOPE  0.2f

typedef __attribute__((ext_vector_type(2))) float v2f;
typedef __attribute__((ext_vector_type(8))) float v8f;

static __device__ __forceinline__ float atomicMaxF(float* addr, float val) {
    int* ai = (int*)addr;
    int old = __float_as_int(*addr);
    while (__int_as_float(old) < val) {
        int assumed = old;
        old = atomicCAS(ai, assumed, __float_as_int(val));
        if (old == assumed) break;
    }
    return __int_as_float(old);
}

// ---------------- utility ----------------
__global__ void fill_f32(float* p, float v, long n) {
    long i = (long)blockIdx.x * blockDim.x + threadIdx.x;
    long s = (long)gridDim.x * blockDim.x;
    for (; i < n; i += s) p[i] = v;
}

__global__ void count_dst(const int* __restrict__ idx, float* __restrict__ cnt, int n) {
    int i = blockIdx.x * blockDim.x + threadIdx.x;
    if (i < n) atomicAdd(&cnt[idx[i]], 1.0f);
}

__global__ void rsqrt_inplace(float* p, int n) {
    int i = blockIdx.x * blockDim.x + threadIdx.x;
    if (i < n) p[i] = rsqrtf(p[i]);
}

// ---------------- WMMA fp32 GEMM: Out[M,Hout] = A[M,K] @ W[K,Hout] (+ A2 @ W2) -----
// wave32: each wave computes one 16x16 f32 tile with V_WMMA_F32_16X16X4_F32.
// Weights staged in LDS (K*Hout <= 8192 floats for every GEMM in this net).
__global__ void wmma_gemm(const float* __restrict__ A, const float* __restrict__ W,
                          float* __restrict__ Out, int M, int K, int Hout,
                          const float* __restrict__ A2, const float* __restrict__ W2) {
    __shared__ float sW[8192];
    const int KH = K * Hout;
    for (int i = threadIdx.x; i < KH; i += blockDim.x) sW[i] = W[i];
    if (A2) {
        for (int i = threadIdx.x; i < KH; i += blockDim.x) sW[KH + i] = W2[i];
    }
    __syncthreads();

    const int lane = threadIdx.x & 31;
    const int wave = threadIdx.x >> 5;
    const int tilesX = Hout >> 4;
    const int totalTiles = (M >> 4) * tilesX;
    const int tile = blockIdx.x * 8 + wave;
    if (tile >= totalTiles) return;

    const int rowBase = (tile / tilesX) << 4;
    const int colBase = (tile % tilesX) << 4;
    const int row = rowBase + (lane & 15);
    const int col = colBase + (lane & 15);
    const int khi = (lane >> 4) << 1;   // lanes 0-15 hold K=k,k+1 ; lanes 16-31 hold K=k+2,k+3

    v8f c = {};
    const float* Arow = A + (long)row * K;
    for (int k = 0; k < K; k += 4) {
        const int kk = k + khi;
        v2f a, b;
        a.x = Arow[kk];
        a.y = Arow[kk + 1];
        b.x = sW[kk * Hout + col];
        b.y = sW[(kk + 1) * Hout + col];
        c = __builtin_amdgcn_wmma_f32_16x16x4_f32(false, a, false, b, (short)0, c, false, false);
    }
    if (A2) {
        const float* A2row = A2 + (long)row * K;
        const float* sW2 = sW + KH;
        for (int k = 0; k < K; k += 4) {
            const int kk = k + khi;
            v2f a, b;
            a.x = A2row[kk];
            a.y = A2row[kk + 1];
            b.x = sW2[kk * Hout + col];
            b.y = sW2[(kk + 1) * Hout + col];
            c = __builtin_amdgcn_wmma_f32_16x16x4_f32(false, a, false, b, (short)0, c, false, false);
        }
    }
    // C/D layout: VGPR i -> M = i (lanes 0-15) or i+8 (lanes 16-31), N = lane&15
    const int mrow = rowBase + ((lane >> 4) << 3);
#pragma unroll
    for (int i = 0; i < 8; ++i)
        Out[(long)(mrow + i) * Hout + colBase + (lane & 15)] = c[i];
}

// ---------------- GCN edge scatter: x1[d] += xw[s] * dinv[s]*dinv[d]  (incl self-loops)
__global__ void gcn_scatter(const float* __restrict__ xw, const int* __restrict__ src,
                            const int* __restrict__ dst, const float* __restrict__ dinv,
                            float* __restrict__ x1, int E, int N) {
    long t = (long)blockIdx.x * blockDim.x + threadIdx.x;
    if (t >= (long)(E + N) * 16) return;
    int e = (int)(t >> 4);
    int q = (int)(t & 15);
    int s = (e < E) ? src[e] : e - E;
    int d = (e < E) ? dst[e] : e - E;
    float nrm = dinv[s] * dinv[d];
    float4 v = *((const float4*)(xw + (long)s * HID) + q);
    float* o = x1 + (long)d * HID + q * 4;
    atomicAdd(o + 0, v.x * nrm);
    atomicAdd(o + 1, v.y * nrm);
    atomicAdd(o + 2, v.z * nrm);
    atomicAdd(o + 3, v.w * nrm);
}

// ---------------- SAGE mean-aggregation scatter ----------------
__global__ void sage_scatter(const float* __restrict__ x1, const int* __restrict__ src,
                             const int* __restrict__ dst, float* __restrict__ agg, int E) {
    long t = (long)blockIdx.x * blockDim.x + threadIdx.x;
    if (t >= (long)E * 16) return;
    int e = (int)(t >> 4);
    int q = (int)(t & 15);
    int s = src[e], d = dst[e];
    float4 v = *((const float4*)(x1 + (long)s * HID) + q);
    float* o = agg + (long)d * HID + q * 4;
    atomicAdd(o + 0, v.x);
    atomicAdd(o + 1, v.y);
    atomicAdd(o + 2, v.z);
    atomicAdd(o + 3, v.w);
}

__global__ void sage_div(float* __restrict__ agg, const float* __restrict__ cnt, int N) {
    long i = (long)blockIdx.x * blockDim.x + threadIdx.x;
    if (i >= (long)N * HID) return;
    int n = (int)(i >> 6);
    agg[i] /= fmaxf(cnt[n], 1.0f);
}

// ---------------- BatchNorm (64 columns): pass 1 sum/sumsq, pass 2 normalize+relu ----
__global__ void bn_stats64(const float* __restrict__ X, float* __restrict__ stats, int M) {
    int c = threadIdx.x & 63;
    int r0 = blockIdx.x * (blockDim.x >> 6) + (threadIdx.x >> 6);
    int rs = gridDim.x * (blockDim.x >> 6);
    float s = 0.f, s2 = 0.f;
    for (int r = r0; r < M; r += rs) {
        float v = X[(long)r * 64 + c];
        s += v;
        s2 += v * v;
    }
    atomicAdd(&stats[c], s);
    atomicAdd(&stats[64 + c], s2);
}

__global__ void bn_apply_relu64(float* __restrict__ X, const float* __restrict__ stats,
                                const float* __restrict__ g, const float* __restrict__ be, int M) {
    long i = (long)blockIdx.x * blockDim.x + threadIdx.x;
    if (i >= (long)M * 64) return;
    int c = (int)(i & 63);
    float mu = stats[c] / (float)M;
    float var = stats[64 + c] / (float)M - mu * mu;
    float v = (X[i] - mu) * rsqrtf(var + EPSV) * g[c] + be[c];
    X[i] = fmaxf(v, 0.f);
}

// ---------------- GAT attention ----------------
__global__ void attn_coef(const float* __restrict__ xg, const float* __restrict__ att_src,
                          const float* __restrict__ att_dst, float* __restrict__ a_s,
                          float* __restrict__ a_d, int N) {
    int t = blockIdx.x * blockDim.x + threadIdx.x;
    if (t >= N * HEADS) return;
    int n = t >> 1, h = t & 1;
    const float* xp = xg + (long)n * (HEADS * HID) + h * HID;
    const float* as = att_src + h * HID;
    const float* ad = att_dst + h * HID;
    float ss = 0.f, sd = 0.f;
    for (int j = 0; j < HID; ++j) {
        float v = xp[j];
        ss += v * as[j];
        sd += v * ad[j];
    }
    a_s[t] = ss;
    a_d[t] = sd;
}

__global__ void attn_max(const float* __restrict__ a_s, const float* __restrict__ a_d,
                         const int* __restrict__ src, const int* __restrict__ dst,
                         float* __restrict__ mbuf, int E, int N) {
    long t = (long)blockIdx.x * blockDim.x + threadIdx.x;
    if (t >= (long)(E + N) * HEADS) return;
    int e = (int)(t >> 1), h = (int)(t & 1);
    int s = (e < E) ? src[e] : e - E;
    int d = (e < E) ? dst[e] : e - E;
    float v = a_s[s * 2 + h] + a_d[d * 2 + h];
    v = (v > 0.f) ? v : NSLOPE * v;
    atomicMaxF(&mbuf[d * 2 + h], v);
}

__global__ void attn_sum(const float* __restrict__ a_s, const float* __restrict__ a_d,
                         const float* __restrict__ mbuf, const int* __restrict__ src,
                         const int* __restrict__ dst, float* __restrict__ ssum, int E, int N) {
    long t = (long)blockIdx.x * blockDim.x + threadIdx.x;
    if (t >= (long)(E + N) * HEADS) return;
    int e = (int)(t >> 1), h = (int)(t & 1);
    int s = (e < E) ? src[e] : e - E;
    int d = (e < E) ? dst[e] : e - E;
    float v = a_s[s * 2 + h] + a_d[d * 2 + h];
    v = (v > 0.f) ? v : NSLOPE * v;
    atomicAdd(&ssum[d * 2 + h], __expf(v - mbuf[d * 2 + h]));
}

// fused: x3[d] += mean_h( alpha_h * xg[s, h, :] )
__global__ void attn_scatter(const float* __restrict__ xg, const float* __restrict__ a_s,
                             const float* __restrict__ a_d, const float* __restrict__ mbuf,
                             const float* __restrict__ ssum, const int* __restrict__ src,
                             const int* __restrict__ dst, float* __restrict__ x3, int E, int N) {
    long t = (long)blockIdx.x * blockDim.x + threadIdx.x;
    if (t >= (long)(E + N) * 16) return;
    int e = (int)(t >> 4);
    int q = (int)(t & 15);
    int s = (e < E) ? src[e] : e - E;
    int d = (e < E) ? dst[e] : e - E;
    float al0, al1;
    {
        float v = a_s[s * 2 + 0] + a_d[d * 2 + 0];
        v = (v > 0.f) ? v : NSLOPE * v;
        al0 = __expf(v - mbuf[d * 2 + 0]) / ssum[d * 2 + 0];
        v = a_s[s * 2 + 1] + a_d[d * 2 + 1];
        v = (v > 0.f) ? v : NSLOPE * v;
        al1 = __expf(v - mbuf[d * 2 + 1]) / ssum[d * 2 + 1];
    }
    float4 v0 = *((const float4*)(xg + (long)s * 128) + q);
    float4 v1 = *((const float4*)(xg + (long)s * 128 + 64) + q);
    float* o = x3 + (long)d * HID + q * 4;
    atomicAdd(o + 0, 0.5f * (al0 * v0.x + al1 * v1.x));
    atomicAdd(o + 1, 0.5f * (al0 * v0.y + al1 * v1.y));
    atomicAdd(o + 2, 0.5f * (al0 * v0.z + al1 * v1.z));
    atomicAdd(o + 3, 0.5f * (al0 * v0.w + al1 * v1.w));
}

// ---------------- pooling + head ----------------
__global__ void pool_sum(const float* __restrict__ x3, const int* __restrict__ batch,
                         float* __restrict__ xsum, int N) {
    long t = (long)blockIdx.x * blockDim.x + threadIdx.x;
    if (t >= (long)N * HID) return;
    int n = (int)(t >> 6), c = (int)(t & 63);
    atomicAdd(&xsum[batch[n] * HID + c], x3[t]);
}

__global__ void build_xc(const float* __restrict__ xsum, const float* __restrict__ nct,
                         float* __restrict__ xc, int B) {
    int t = blockIdx.x * blockDim.x + threadIdx.x;
    if (t >= B * HID) return;
    int b = t >> 6, c = t & 63;
    float s = xsum[b * HID + c];
    xc[b * 128 + c] = s / nct[b];
    xc[b * 128 + 64 + c] = s;
}

__global__ void mlp_out(const float* __restrict__ t1, const float* __restrict__ W,
                        const float* __restrict__ b, float* __restrict__ out, int B) {
    int t = blockIdx.x * blockDim.x + threadIdx.x;
    if (t >= B * OUT_DIM) return;
    int bi = t / OUT_DIM, o = t % OUT_DIM;
    float acc = b[o];
    for (int j = 0; j < HID; ++j) acc += t1[bi * HID + j] * W[j * OUT_DIM + o];
    out[t] = acc;
}

// =========================================================================
extern "C" void kernel_launch(void* const* d_in, const int* in_sizes, int n_in,
                              void* d_out, int out_size, void* d_ws, size_t ws_size,
                              hipStream_t stream) {
    const float* x       = (const float*)d_in[0];
    const int*   ei      = (const int*)d_in[1];
    const int*   src     = ei;
    const int*   dst     = ei + N_EDGES;
    const int*   batch   = (const int*)d_in[2];
    const float* W_gcn   = (const float*)d_in[3];
    const float* W_sl    = (const float*)d_in[5];
    const float* W_sr    = (const float*)d_in[6];
    const float* W_gat   = (const float*)d_in[8];
    const float* att_src = (const float*)d_in[9];
    const float* att_dst = (const float*)d_in[10];
    const float* g1 = (const float*)d_in[12], *be1 = (const float*)d_in[13];
    const float* g2 = (const float*)d_in[14], *be2 = (const float*)d_in[15];
    const float* g3 = (const float*)d_in[16], *be3 = (const float*)d_in[17];
    const float* W_m1 = (const float*)d_in[18];
    const float* gm = (const float*)d_in[20], *bem = (const float*)d_in[21];
    const float* W_m2 = (const float*)d_in[22], *b_m2 = (const float*)d_in[23];
    float* out = (float*)d_out;

    // ---- workspace carve-up ----
    char* ws = (char*)d_ws;
    size_t off = 0;
    auto carve = [&](size_t bytes) -> float* {
        float* p = (float*)(ws + off);
        off += (bytes + 255) & ~(size_t)255;
        return p;
    };
    const size_t N = N_NODES, E = N_EDGES;
    float* deg   = carve(N * 4);          // -> dinv (in place)
    float* cnt   = carve(N * 4);
    float* a_s   = carve(N * 2 * 4);
    float* a_d   = carve(N * 2 * 4);
    float* mbuf  = carve(N * 2 * 4);
    float* ssum  = carve(N * 2 * 4);
    float* stats = carve(128 * 4);
    float* nct   = carve(B_GR * 4);
    float* xsum  = carve(B_GR * HID * 4);
    float* xc    = carve(B_GR * 128 * 4);
    float* t1    = carve(B_GR * HID * 4);
    float* bufA  = carve(N * HID * 4);    // xw -> agg -> x3
    float* bufB  = carve(N * HID * 4);    // x1
    float* bufC  = carve(N * HID * 4);    // x2
    float* bufD  = carve(N * 128 * 4);    // xg
    (void)ws_size; (void)in_sizes; (void)n_in; (void)out_size;

    const int T = 256;
    auto cdiv = [](long a, long b) { return (int)((a + b - 1) / b); };
    const long NH = (long)N * HID;
    const long ESL16 = (long)(E + N) * 16;

    // ================= GCN layer =================
    fill_f32<<<512, T, 0, stream>>>(bufB, 0.f, NH);                 // x1 accumulator
    fill_f32<<<512, T, 0, stream>>>(deg, 1.f, (long)N);             // self-loop contributes 1
    count_dst<<<cdiv(E, T), T, 0, stream>>>(dst, deg, (int)E);
    rsqrt_inplace<<<cdiv(N, T), T, 0, stream>>>(deg, (int)N);
    // xw = x @ W_gcn   [100000x128 @ 128x64]
    wmma_gemm<<<cdiv((N / 16) * (HID / 16), 8), T, 0, stream>>>(
        x, W_gcn, bufA, (int)N, C_IN, HID, nullptr, nullptr);
    gcn_scatter<<<cdiv(ESL16, T), T, 0, stream>>>(bufA, src, dst, deg, bufB, (int)E, (int)N);
    fill_f32<<<1, 128, 0, stream>>>(stats, 0.f, 128);
    bn_stats64<<<256, T, 0, stream>>>(bufB, stats, (int)N);
    bn_apply_relu64<<<cdiv(NH, T), T, 0, stream>>>(bufB, stats, g1, be1, (int)N);

    // ================= SAGE layer =================
    fill_f32<<<cdiv(N, T), T, 0, stream>>>(cnt, 0.f, (long)N);
    count_dst<<<cdiv(E, T), T, 0, stream>>>(dst, cnt, (int)E);
    fill_f32<<<512, T, 0, stream>>>(bufA, 0.f, NH);                 // agg
    sage_scatter<<<cdiv((long)E * 16, T), T, 0, stream>>>(bufB, src, dst, bufA, (int)E);
    sage_div<<<cdiv(NH, T), T, 0, stream>>>(bufA, cnt, (int)N);
    // x2 = agg @ W_sl + x1 @ W_sr   (fused dual GEMM, K=64, Hout=64)
    wmma_gemm<<<cdiv((N / 16) * (HID / 16), 8), T, 0, stream>>>(
        bufA, W_sl, bufC, (int)N, HID, HID, bufB, W_sr);
    fill_f32<<<1, 128, 0, stream>>>(stats, 0.f, 128);
    bn_stats64<<<256, T, 0, stream>>>(bufC, stats, (int)N);
    bn_apply_relu64<<<cdiv(NH, T), T, 0, stream>>>(bufC, stats, g2, be2, (int)N);

    // ================= GAT layer =================
    // xg = x2 @ W_gat   [100000x64 @ 64x128]
    wmma_gemm<<<cdiv((N / 16) * (128 / 16), 8), T, 0, stream>>>(
        bufC, W_gat, bufD, (int)N, HID, 128, nullptr, nullptr);
    attn_coef<<<cdiv((long)N * HEADS, T), T, 0, stream>>>(bufD, att_src, att_dst, a_s, a_d, (int)N);
    fill_f32<<<cdiv((long)N * 2, T), T, 0, stream>>>(mbuf, -1e30f, (long)N * 2);
    attn_max<<<cdiv((long)(E + N) * 2, T), T, 0, stream>>>(a_s, a_d, src, dst, mbuf, (int)E, (int)N);
    fill_f32<<<cdiv((long)N * 2, T), T, 0, stream>>>(ssum, 0.f, (long)N * 2);
    attn_sum<<<cdiv((long)(E + N) * 2, T), T, 0, stream>>>(a_s, a_d, mbuf, src, dst, ssum, (int)E, (int)N);
    fill_f32<<<512, T, 0, stream>>>(bufA, 0.f, NH);                 // x3
    attn_scatter<<<cdiv(ESL16, T), T, 0, stream>>>(bufD, a_s, a_d, mbuf, ssum, src, dst,
                                                   bufA, (int)E, (int)N);
    fill_f32<<<1, 128, 0, stream>>>(stats, 0.f, 128);
    bn_stats64<<<256, T, 0, stream>>>(bufA, stats, (int)N);
    bn_apply_relu64<<<cdiv(NH, T), T, 0, stream>>>(bufA, stats, g3, be3, (int)N);

    // ================= pooling =================
    fill_f32<<<1, B_GR, 0, stream>>>(nct, 0.f, B_GR);
    count_dst<<<cdiv(N, T), T, 0, stream>>>(batch, nct, (int)N);
    fill_f32<<<cdiv((long)B_GR * HID, T), T, 0, stream>>>(xsum, 0.f, (long)B_GR * HID);
    pool_sum<<<cdiv(NH, T), T, 0, stream>>>(bufA, batch, xsum, (int)N);
    build_xc<<<cdiv((long)B_GR * HID, T), T, 0, stream>>>(xsum, nct, xc, B_GR);

    // ================= MLP head =================
    // t1 = xc @ W_m1   [64x128 @ 128x64]
    wmma_gemm<<<cdiv((B_GR / 16) * (HID / 16), 8), T, 0, stream>>>(
        xc, W_m1, t1, B_GR, 128, HID, nullptr, nullptr);
    fill_f32<<<1, 128, 0, stream>>>(stats, 0.f, 128);
    bn_stats64<<<1, T, 0, stream>>>(t1, stats, B_GR);
    bn_apply_relu64<<<cdiv((long)B_GR * HID, T), T, 0, stream>>>(t1, stats, gm, bem, B_GR);
    mlp_out<<<cdiv((long)B_GR * OUT_DIM, T), T, 0, stream>>>(t1, W_m2, b_m2, out, B_GR);
}